// MAAttentionLayer_23210003267967
// MI455X (gfx1250) — compile-verified
//
#include <hip/hip_runtime.h>

typedef __attribute__((ext_vector_type(2))) float v2f;
typedef __attribute__((ext_vector_type(8))) float v8f;

#define IN_DIM 256
#define M_ROWS 4096
#define A_ROWS 12288
#define N_ROWS (M_ROWS + A_ROWS) /* 16384 */
#define CS_BLOCKS 128
#define ROWS_PER_BLOCK (N_ROWS / CS_BLOCKS) /* 128 */

// ---------------- Stage A: partial column sums of combine_new ----------------
// partial[b*256 + col] = sum over this block's 128 rows of combine_new[row][col]
__global__ void __launch_bounds__(IN_DIM)
maatt_colsum_partial(const float* __restrict__ mashup,
                     const float* __restrict__ new_api,
                     float* __restrict__ partial) {
    const int col  = threadIdx.x;            // 0..255 -> coalesced across lanes
    const int row0 = blockIdx.x * ROWS_PER_BLOCK;
    float acc = 0.0f;
#pragma unroll 4
    for (int r = 0; r < ROWS_PER_BLOCK; ++r) {
        const int row = row0 + r;
        const float* src = (row < M_ROWS)
            ? (mashup  + (size_t)row * IN_DIM)
            : (new_api + (size_t)(row - M_ROWS) * IN_DIM);
        acc += src[col];
    }
    partial[blockIdx.x * IN_DIM + col] = acc;
}

// ---------------- Stage B: colsum -> s -> t, c (all tiny) ----------------
// s = colsum @ k_w + N*k_b ; t = q_w @ s ; c = q_b . s
__global__ void __launch_bounds__(IN_DIM)
maatt_small_solve(const float* __restrict__ partial,
                  const float* __restrict__ q_w,
                  const float* __restrict__ q_b,
                  const float* __restrict__ k_w,
                  const float* __restrict__ k_b,
                  float* __restrict__ t_out,   // [256]
                  float* __restrict__ c_out) { // [1]
    __shared__ float cs[IN_DIM];
    __shared__ float sv[IN_DIM];
    const int tid = threadIdx.x; // 256 threads

    float a = 0.0f;
    for (int b = 0; b < CS_BLOCKS; ++b) a += partial[b * IN_DIM + tid]; // coalesced
    cs[tid] = a;
    __syncthreads();

    float s = (float)N_ROWS * k_b[tid];
    for (int d = 0; d < IN_DIM; ++d) s += cs[d] * k_w[d * IN_DIM + tid]; // coalesced
    sv[tid] = s;
    __syncthreads();

    float t = 0.0f;
    for (int o = 0; o < IN_DIM; ++o) t += q_w[tid * IN_DIM + o] * sv[o];
    t_out[tid] = t;

    if (tid == 0) {
        float c = 0.0f;
        for (int o = 0; o < IN_DIM; ++o) c += q_b[o] * sv[o];
        *c_out = c;
    }
}

// ---------------- Stage C: out[i] = (combine[i] . t + c) / 16 via WMMA ----------------
// One wave32 per 16-row tile. A = 16x4 fp32 chunk of combine (documented layout:
// lane 0-15 = M, lane-half/VGPR selects K). B = t[k0..k0+3] broadcast across all
// 16 N columns, so every C column holds the same per-row dot product.
__global__ void __launch_bounds__(256)
maatt_dot_wmma(const float* __restrict__ mashup,
               const float* __restrict__ api,
               const float* __restrict__ t_vec,
               const float* __restrict__ c_scalar,
               float* __restrict__ out) {
    __shared__ float tls[IN_DIM];
    __shared__ float csh;
    const int tid = threadIdx.x;
    tls[tid & (IN_DIM - 1)] = t_vec[tid & (IN_DIM - 1)];
    if (tid == 0) csh = *c_scalar;
    __syncthreads();

    const int wave = tid >> 5;                 // 8 waves per block
    const int lane = tid & 31;
    const int base = (blockIdx.x * 8 + wave) * 16; // tile's first row
    const int m    = lane & 15;                // A: M index
    const int koff = (lane >> 4) << 1;         // A: K sub-offset 0 or 2

    const size_t row = (size_t)(base + m);
    const float* rp = (row < M_ROWS)
        ? (mashup + row * (size_t)IN_DIM)
        : (api + (row - M_ROWS) * (size_t)IN_DIM);

    v8f acc = {0.f, 0.f, 0.f, 0.f, 0.f, 0.f, 0.f, 0.f};
#pragma unroll 4
    for (int k0 = 0; k0 < IN_DIM; k0 += 4) {
        // A[m][k0+koff], A[m][k0+koff+1]  (VGPR0=K0/K2, VGPR1=K1/K3 per lane-half)
        v2f a = *(const v2f*)(rp + k0 + koff);      // global_load_b64, 8B aligned
        v2f b = *(const v2f*)(&tls[k0 + koff]);     // ds_load_b64 broadcast
        acc = __builtin_amdgcn_wmma_f32_16x16x4_f32(
            /*neg_a=*/false, a, /*neg_b=*/false, b,
            /*c_mod=*/(short)0, acc, /*reuse_a=*/false, /*reuse_b=*/false);
    }

    const float c = csh;
    // C layout: lane 0 VGPR r = (M=r, N=0), lane 16 VGPR r = (M=8+r, N=0);
    // all N columns identical because B was broadcast.
    if ((lane & 15) == 0) {
        const int mbase = base + ((lane >> 4) << 3);
#pragma unroll
        for (int r = 0; r < 8; ++r)
            out[mbase + r] = (acc[r] + c) * 0.0625f; // 1/sqrt(256)
    }
}

extern "C" void kernel_launch(void* const* d_in, const int* in_sizes, int n_in,
                              void* d_out, int out_size, void* d_ws, size_t ws_size,
                              hipStream_t stream) {
    const float* mashup  = (const float*)d_in[0];
    const float* api     = (const float*)d_in[1];
    const float* new_api = (const float*)d_in[2];
    const float* q_w     = (const float*)d_in[3];
    const float* q_b     = (const float*)d_in[4];
    const float* k_w     = (const float*)d_in[5];
    const float* k_b     = (const float*)d_in[6];
    float* out = (float*)d_out;

    float* ws      = (float*)d_ws;
    float* partial = ws;                          // 128*256 floats
    float* t_vec   = ws + CS_BLOCKS * IN_DIM;     // 256 floats
    float* c_s     = t_vec + IN_DIM;              // 1 float

    maatt_colsum_partial<<<CS_BLOCKS, IN_DIM, 0, stream>>>(mashup, new_api, partial);
    maatt_small_solve<<<1, IN_DIM, 0, stream>>>(partial, q_w, q_b, k_w, k_b, t_vec, c_s);
    // 16384/16 = 1024 tiles, 8 waves (tiles) per 256-thread block -> 128 blocks
    maatt_dot_wmma<<<N_ROWS / 16 / 8, 256, 0, stream>>>(mashup, api, t_vec, c_s, out);
}